// TTTMoE_1821066134034
// MI455X (gfx1250) — compile-verified
//
#include <hip/hip_runtime.h>

typedef __attribute__((ext_vector_type(2))) float v2f;
typedef __attribute__((ext_vector_type(8))) float v8f;

#define TB 64
#define TN 8192
#define TD 64
#define TH 128
#define CHUNKS 16
#define RPI 32
#define ITERS ((TN / CHUNKS) / RPI)   // 16
#define BLOCK 256
#define BDH (TB * TD * TH)            // 524288
#define ACCN (2 * BDH + 2 * TB)       // G0 | G1 | S[b] | Lsum[b]

__device__ __forceinline__ v8f vzero8() {
  v8f v;
#pragma unroll
  for (int i = 0; i < 8; ++i) v[i] = 0.0f;
  return v;
}

__device__ __forceinline__ v8f wmma4(v2f a, v2f b, v8f c) {
  // D = A(16x4,f32) x B(4x16,f32) + C(16x16,f32)
  return __builtin_amdgcn_wmma_f32_16x16x4_f32(false, a, false, b, (short)0, c,
                                               false, false);
}

__device__ __forceinline__ float gelu_f(float x) {
  return 0.5f * x * (1.0f + erff(x * 0.7071067811865476f));
}
__device__ __forceinline__ float gelu_df(float x) {
  float phi = 0.5f * (1.0f + erff(x * 0.7071067811865476f));
  return phi + x * 0.3989422804014327f * __expf(-0.5f * x * x);
}

// A fragment: row-major M, 16x4 block at (r0,k0), leading-dim ld.
// ISA layout: VGPR0 = {K0 lanes0-15, K2 lanes16-31}, VGPR1 = {K1, K3}.
__device__ __forceinline__ v2f fragA(const float* m, int ld, int r0, int k0,
                                     int l, int half) {
  const float* p = m + (r0 + l) * ld + k0 + 2 * half;
  v2f a; a.x = p[0]; a.y = p[1]; return a;
}
// B fragment: row-major M, 4x16 block at (k0,c0). Also serves as A^T fragment.
__device__ __forceinline__ v2f fragB(const float* m, int ld, int k0, int c0,
                                     int l, int half) {
  const float* p = m + (k0 + 2 * half) * ld + c0 + l;
  v2f b; b.x = p[0]; b.y = p[ld]; return b;
}
// B = W^T with W row-major: B[k][c] = W[c][k]
__device__ __forceinline__ v2f fragBT(const float* w, int ld, int k0, int c0,
                                      int l, int half) {
  const float* p = w + (c0 + l) * ld + k0 + 2 * half;
  v2f b; b.x = p[0]; b.y = p[1]; return b;
}

// CDNA5 async Global->LDS DMA (ASYNCcnt-tracked, VGPR-bypassing).
// lds_off: byte offset in LDS space (generic-pointer low 32 bits).
__device__ __forceinline__ void async_ld_b128(unsigned lds_off,
                                              const float4* g) {
  asm volatile("global_load_async_to_lds_b128 %0, %1, off"
               :: "v"(lds_off), "v"(g)
               : "memory");
}

__global__ void ttt_zero(float* acc) {
  int i = blockIdx.x * blockDim.x + threadIdx.x;
  if (i < ACCN) acc[i] = 0.0f;
}

__global__ __launch_bounds__(BLOCK, 1) void ttt_main(
    const float* __restrict__ X, const float* __restrict__ w0,
    const float* __restrict__ w1, const float* __restrict__ wk,
    float* __restrict__ acc) {
  __shared__ __align__(16) float s_wk[TD * TD];        // 16 KB
  __shared__ __align__(16) float s_w0[TD * TH];        // 32 KB
  __shared__ __align__(16) float s_w1[TH * TD];        // 32 KB
  __shared__ __align__(16) float s_Xbuf[2][RPI * TD];  // 16 KB ping-pong X
  __shared__ __align__(16) float s_R[RPI * TD];        // 8 KB residual
  __shared__ __align__(16) float s_k[RPI * TD];        // 8 KB
  __shared__ __align__(16) float s_h[RPI * TH];        // 16 KB pre-activation
  __shared__ __align__(16) float s_dH[RPI * TH];       // 16 KB
  __shared__ float s_red[BLOCK];

  const int b = blockIdx.x / CHUNKS;
  const int chunk = blockIdx.x % CHUNKS;
  const int tid = threadIdx.x;
  const int wid = tid >> 5;      // 8 wave32 per block
  const int lane = tid & 31;
  const int half = lane >> 4;
  const int l = lane & 15;

  const float* Xb =
      X + (size_t)b * TN * TD + (size_t)chunk * (TN / CHUNKS) * TD;

  // Issue async DMA of X tile `tile` into ping-pong buffer `bufidx`.
  // RPI*TD floats = 512 float4 => 2 x b128 per thread, asynccnt += 2 per wave.
  auto issue_tile = [&](int tile, int bufidx) {
    const float4* src = (const float4*)(Xb + (size_t)tile * RPI * TD);
    unsigned base = (unsigned)(uintptr_t)(void*)&s_Xbuf[bufidx][0];
    async_ld_b128(base + (unsigned)tid * 16u, src + tid);
    async_ld_b128(base + (unsigned)(tid + BLOCK) * 16u, src + tid + BLOCK);
  };

  issue_tile(0, 0);  // prime the pipeline

  for (int i = tid; i < TD * TD; i += BLOCK) s_wk[i] = wk[i];
  for (int i = tid; i < TD * TH; i += BLOCK) s_w0[i] = w0[i];
  for (int i = tid; i < TH * TD; i += BLOCK) s_w1[i] = w1[i];

  v8f accG0[4], accG1[4];
#pragma unroll
  for (int t = 0; t < 4; ++t) { accG0[t] = vzero8(); accG1[t] = vzero8(); }
  float sum_k = 0.0f, sum_l = 0.0f;

  __syncthreads();

  int cur = 0;
  for (int it = 0; it < ITERS; ++it) {
    // ---- pipeline: start DMA of next tile, wait for current tile ----
    if (it + 1 < ITERS) {
      issue_tile(it + 1, cur ^ 1);  // buf was fully consumed 5 barriers ago
      // async loads complete in order: <=2 outstanding means current tile done
      asm volatile("s_wait_asynccnt 0x2" ::: "memory");
    } else {
      asm volatile("s_wait_asynccnt 0x0" ::: "memory");
    }
    __syncthreads();
    const float* s_X = s_Xbuf[cur];

    // ---- k = X @ wk^T : 2x4 tiles, 1 tile per wave ----
    {
      const int rt = (wid >> 2) * 16, ct = (wid & 3) * 16;
      v8f c = vzero8();
#pragma unroll
      for (int kk = 0; kk < TD; kk += 4)
        c = wmma4(fragA(s_X, TD, rt, kk, l, half),
                  fragBT(s_wk, TD, kk, ct, l, half), c);
#pragma unroll
      for (int r = 0; r < 8; ++r) {
        s_k[(rt + r + 8 * half) * TD + ct + l] = c[r];
        sum_k += c[r];
      }
    }
    __syncthreads();

    // ---- h = k @ w0 : 2x8 tiles, 2 per wave ----
#pragma unroll
    for (int tt = 0; tt < 2; ++tt) {
      const int t = wid * 2 + tt;
      const int rt = (t >> 3) * 16, ct = (t & 7) * 16;
      v8f c = vzero8();
#pragma unroll
      for (int kk = 0; kk < TD; kk += 4)
        c = wmma4(fragA(s_k, TD, rt, kk, l, half),
                  fragB(s_w0, TH, kk, ct, l, half), c);
#pragma unroll
      for (int r = 0; r < 8; ++r)
        s_h[(rt + r + 8 * half) * TH + ct + l] = c[r];
    }
    __syncthreads();

    // ---- R = gelu(h) @ w1 - k : 2x4 tiles, 1 per wave ----
    {
      const int rt = (wid >> 2) * 16, ct = (wid & 3) * 16;
      v8f c = vzero8();
#pragma unroll
      for (int kk = 0; kk < TH; kk += 4) {
        const float* p = s_h + (rt + l) * TH + kk + 2 * half;
        v2f a; a.x = gelu_f(p[0]); a.y = gelu_f(p[1]);
        c = wmma4(a, fragB(s_w1, TD, kk, ct, l, half), c);
      }
#pragma unroll
      for (int r = 0; r < 8; ++r) {
        const int rr = rt + r + 8 * half, cc = ct + l;
        float Rv = c[r] - s_k[rr * TD + cc];
        s_R[rr * TD + cc] = Rv;
        sum_l += Rv * Rv;
      }
    }
    __syncthreads();

    // ---- dH = (R @ w1^T) * gelu'(h) : 2x8 tiles, 2 per wave ----
#pragma unroll
    for (int tt = 0; tt < 2; ++tt) {
      const int t = wid * 2 + tt;
      const int rt = (t >> 3) * 16, ct = (t & 7) * 16;
      v8f c = vzero8();
#pragma unroll
      for (int kk = 0; kk < TD; kk += 4)
        c = wmma4(fragA(s_R, TD, rt, kk, l, half),
                  fragBT(s_w1, TD, kk, ct, l, half), c);
#pragma unroll
      for (int r = 0; r < 8; ++r) {
        const int rr = rt + r + 8 * half, cc = ct + l;
        s_dH[rr * TH + cc] = c[r] * gelu_df(s_h[rr * TH + cc]);
      }
    }
    __syncthreads();

    // ---- G1 += g^T @ R : 8x4 output tiles, 4 per wave (K = rows) ----
#pragma unroll
    for (int tt = 0; tt < 4; ++tt) {
      const int t = wid * 4 + tt;
      const int ht = (t >> 2) * 16, jt = (t & 3) * 16;
      v8f c = accG1[tt];
#pragma unroll
      for (int kk = 0; kk < RPI; kk += 4) {
        const float* p = s_h + (kk + 2 * half) * TH + ht + l;  // g^T fragment
        v2f a; a.x = gelu_f(p[0]); a.y = gelu_f(p[TH]);
        c = wmma4(a, fragB(s_R, TD, kk, jt, l, half), c);
      }
      accG1[tt] = c;
    }
    // ---- G0 += k^T @ dH : 4x8 output tiles, 4 per wave ----
#pragma unroll
    for (int tt = 0; tt < 4; ++tt) {
      const int t = wid * 4 + tt;
      const int dt = (t >> 3) * 16, jt = (t & 7) * 16;
      v8f c = accG0[tt];
#pragma unroll
      for (int kk = 0; kk < RPI; kk += 4)
        c = wmma4(fragB(s_k, TD, kk, dt, l, half),   // k^T fragment
                  fragB(s_dH, TH, kk, jt, l, half), c);
      accG0[tt] = c;
    }
    __syncthreads();
    cur ^= 1;
  }

  // ---- merge per-workgroup partials via fp32 global atomics ----
  float* G0b = acc + (size_t)b * TD * TH;
  float* G1b = acc + (size_t)BDH + (size_t)b * TH * TD;
#pragma unroll
  for (int tt = 0; tt < 4; ++tt) {
    const int t = wid * 4 + tt;
    const int dt = (t >> 3) * 16, jt = (t & 7) * 16;
#pragma unroll
    for (int r = 0; r < 8; ++r)
      unsafeAtomicAdd(&G0b[(dt + r + 8 * half) * TH + jt + l], accG0[tt][r]);
  }
#pragma unroll
  for (int tt = 0; tt < 4; ++tt) {
    const int t = wid * 4 + tt;
    const int ht = (t >> 2) * 16, jt = (t & 3) * 16;
#pragma unroll
    for (int r = 0; r < 8; ++r)
      unsafeAtomicAdd(&G1b[(ht + r + 8 * half) * TD + jt + l], accG1[tt][r]);
  }

  s_red[tid] = sum_k;
  __syncthreads();
  for (int off = BLOCK / 2; off > 0; off >>= 1) {
    if (tid < off) s_red[tid] += s_red[tid + off];
    __syncthreads();
  }
  if (tid == 0) unsafeAtomicAdd(&acc[2 * BDH + b], s_red[0]);
  __syncthreads();
  s_red[tid] = sum_l;
  __syncthreads();
  for (int off = BLOCK / 2; off > 0; off >>= 1) {
    if (tid < off) s_red[tid] += s_red[tid + off];
    __syncthreads();
  }
  if (tid == 0) unsafeAtomicAdd(&acc[2 * BDH + TB + b], s_red[0]);
}

__global__ void ttt_finalize(const float* __restrict__ acc,
                             float* __restrict__ out) {
  const float inv = 1.0f / ((float)TN * (float)TD);
  int i = blockIdx.x * blockDim.x + threadIdx.x;
  const int total = 2 * BDH + TB;
  if (i >= total) return;
  if (i < 2 * BDH) {
    int b = (i < BDH) ? (i / (TD * TH)) : ((i - BDH) / (TH * TD));
    float alpha = 2.0f * acc[2 * BDH + b] * inv;  // 2*S/(N*D)
    out[i] = alpha * acc[i];
  } else {
    int b = i - 2 * BDH;
    out[i] = acc[2 * BDH + TB + b] * inv;         // loss = sumR2/(N*D)
  }
}

extern "C" void kernel_launch(void* const* d_in, const int* in_sizes, int n_in,
                              void* d_out, int out_size, void* d_ws,
                              size_t ws_size, hipStream_t stream) {
  const float* X  = (const float*)d_in[0];
  const float* w0 = (const float*)d_in[1];
  const float* w1 = (const float*)d_in[2];
  const float* wk = (const float*)d_in[3];
  float* out = (float*)d_out;
  float* acc = (float*)d_ws;  // needs (2*BDH + 128) * 4 bytes ~= 4 MB
  (void)in_sizes; (void)n_in; (void)out_size; (void)ws_size;

  ttt_zero<<<(ACCN + 255) / 256, 256, 0, stream>>>(acc);
  ttt_main<<<dim3(TB * CHUNKS), dim3(BLOCK), 0, stream>>>(X, w0, w1, wk, acc);
  ttt_finalize<<<(2 * BDH + TB + 255) / 256, 256, 0, stream>>>(acc, out);
}